// GroupEmbedding_72980084294362
// MI455X (gfx1250) — compile-verified
//
#include <hip/hip_runtime.h>

typedef __attribute__((ext_vector_type(2))) float v2f;
typedef __attribute__((ext_vector_type(8))) float v8f;

#define NG 4096   // groups
#define NU 50     // users per group
#define NM 20     // behaviors per user
#define ND 64     // embedding dim
#define NS 64     // similarity dim
#define UPAD 64   // users padded (zero rows beyond NU)
#define PSTRIDE 80 // LDS row stride (floats); 2-row offset = 160 mod 64 banks = 32 -> conflict-free transposed reads
#define FACTOR 0.5f

__global__ __launch_bounds__(256) void GroupEmbedding_72980084294362_kernel(
    const int*   __restrict__ group_user,       // [G,U]
    const int*   __restrict__ behavior_ids,     // [G,U,M]
    const float* __restrict__ behavior_counts,  // [G,U,M]
    const int*   __restrict__ target_user,      // [G]
    const float* __restrict__ similarity_vec,   // [USER_NUM,S]
    const float* __restrict__ user_emb_w,       // [USER_NUM,D]
    const float* __restrict__ item_emb_w,       // [ITEM_NUM,D]
    float*       __restrict__ out)              // [G,D]
{
    __shared__ float s_sim[UPAD];
    __shared__ float s_pers[UPAD][PSTRIDE];

    const int g    = blockIdx.x;
    const int tid  = threadIdx.x;
    const int wave = tid >> 5;
    const int lane = tid & 31;

    // Target similarity vector: 2 consecutive elems per lane (replicated per wave).
    const int tg = target_user[g];
    const float2 tsv = *(const float2*)(similarity_vec + (size_t)tg * NS + 2 * lane);

    // Phases A+B: per user slot -> sim weight and personalized embedding into LDS.
    for (int u = wave; u < UPAD; u += 8) {
        if (u < NU) {
            const int uid = group_user[g * NU + u];

            // --- sim[u] = FACTOR * dot(target_sim, sim_vec[uid]) over S=64 ---
            const float2 sv = *(const float2*)(similarity_vec + (size_t)uid * NS + 2 * lane);
            float dot = sv.x * tsv.x + sv.y * tsv.y;
            #pragma unroll
            for (int o = 16; o > 0; o >>= 1) dot += __shfl_xor(dot, o, 32);
            if (lane == 0) s_sim[u] = FACTOR * dot;

            // --- weighted sum of 20 gathered item rows (coalesced float2/lane) ---
            const int*   ids  = behavior_ids    + ((size_t)g * NU + u) * NM;
            const float* cnts = behavior_counts + ((size_t)g * NU + u) * NM;
            float ax = 0.f, ay = 0.f;
            int id_next = ids[0];
            #pragma unroll 4
            for (int m = 0; m < NM; ++m) {
                const int   id = id_next;
                const float c  = cnts[m];
                if (m + 1 < NM) {
                    id_next = ids[m + 1];
                    // hide L2-gather latency: prefetch the next behavior's row
                    __builtin_prefetch(item_emb_w + (size_t)id_next * ND + 2 * lane, 0, 3);
                }
                const float2 e = *(const float2*)(item_emb_w + (size_t)id * ND + 2 * lane);
                ax = fmaf(c, e.x, ax);
                ay = fmaf(c, e.y, ay);
            }
            const float2 ue = *(const float2*)(user_emb_w + (size_t)uid * ND + 2 * lane);
            s_pers[u][2 * lane]     = ax * ue.x;
            s_pers[u][2 * lane + 1] = ay * ue.y;
        } else {
            // zero padding rows so WMMA K-chunks past NU contribute nothing
            if (lane == 0) s_sim[u] = 0.f;
            s_pers[u][2 * lane]     = 0.f;
            s_pers[u][2 * lane + 1] = 0.f;
        }
    }

    __syncthreads();

    // Phase C: out[g,d] = sum_u sim[u] * pers[u,d] via V_WMMA_F32_16X16X4_F32.
    // A[i,k] = pers[u0+k][d0+i] (16x4 tile of transposed personalized),
    // B[k,j] = sim[u0+k] (replicated along N) -> D[i,j] = out[d0+i] for all j.
    if (wave < 4) {
        const int d0    = wave * 16;
        const int kBase = (lane < 16) ? 0 : 2;  // A/B VGPR0 holds K=0 (lanes 0-15), K=2 (lanes 16-31)
        const int mi    = lane & 15;            // A row index M within tile
        v8f c = {0.f, 0.f, 0.f, 0.f, 0.f, 0.f, 0.f, 0.f};
        for (int u0 = 0; u0 < 52; u0 += 4) {    // ceil(50/4)=13 K-chunks; rows 50,51 are zero
            v2f a, b;
            a.x = s_pers[u0 + kBase    ][d0 + mi];
            a.y = s_pers[u0 + kBase + 1][d0 + mi];
            b.x = s_sim[u0 + kBase];
            b.y = s_sim[u0 + kBase + 1];
            c = __builtin_amdgcn_wmma_f32_16x16x4_f32(
                    /*neg_a=*/false, a, /*neg_b=*/false, b,
                    /*c_mod=*/(short)0, c, /*reuse_a=*/false, /*reuse_b=*/false);
        }
        // Extract column N=0: lane 0 holds M=0..7 in c[0..7], lane 16 holds M=8..15.
        float* og = out + (size_t)g * ND + d0;
        if (lane == 0) {
            #pragma unroll
            for (int r = 0; r < 8; ++r) og[r] = c[r];
        } else if (lane == 16) {
            #pragma unroll
            for (int r = 0; r < 8; ++r) og[8 + r] = c[r];
        }
    }
}

extern "C" void kernel_launch(void* const* d_in, const int* in_sizes, int n_in,
                              void* d_out, int out_size, void* d_ws, size_t ws_size,
                              hipStream_t stream) {
    (void)in_sizes; (void)n_in; (void)d_ws; (void)ws_size; (void)out_size;
    const int*   group_user      = (const int*)  d_in[0];
    const int*   behavior_ids    = (const int*)  d_in[1];
    const float* behavior_counts = (const float*)d_in[2];
    const int*   target_user     = (const int*)  d_in[3];
    const float* similarity_vec  = (const float*)d_in[4];
    const float* user_emb_w      = (const float*)d_in[5];
    const float* item_emb_w      = (const float*)d_in[6];
    float*       out             = (float*)      d_out;

    GroupEmbedding_72980084294362_kernel<<<NG, 256, 0, stream>>>(
        group_user, behavior_ids, behavior_counts, target_user,
        similarity_vec, user_emb_w, item_emb_w, out);
}